// QCNN_Shared_29978871726316
// MI455X (gfx1250) — compile-verified
//
#include <hip/hip_runtime.h>
#include <math.h>

typedef __attribute__((ext_vector_type(2))) float v2f;
typedef __attribute__((ext_vector_type(8))) float v8f;

// XOR-swap lane exchange via ds_swizzle_b32 with an immediate pattern:
// group-of-32 mode, and_mask=0x1F, xor_mask=M, or_mask=0  ->  imm = (M<<5)|0x1F.
// No VALU index setup (unlike ds_bpermute-based __shfl_xor).
template<int M>
__device__ __forceinline__ float swzx(float v) {
  constexpr int IMM = (M << 5) | 0x1F;
  int r = __builtin_amdgcn_ds_swizzle(__builtin_bit_cast(int, v), IMM);
  return __builtin_bit_cast(float, r);
}

// -----------------------------------------------------------------------------
// State layout (one wave32 per batch element):
//   amplitude index = 8 bits, bit(wire w) = 7 - w  (wire 0 is MSB, as in the
//   reference's (B,2,2,...,2) flattening).
//   index bits [7:3] = lane[4:0]  -> wires 0..4 are "lane wires" (ds_swizzle)
//   index bits [2:0] = local idx  -> wires 5..7 are "local wires" (registers)
// Each lane holds 8 complex amplitudes: pr[8], pi[8].  16 VGPRs of state, no
// LDS storage, no barriers, zero spills (vmem stays ~15).
// -----------------------------------------------------------------------------

// Generic (optionally controlled) 1-qubit gate, complex 2x2 U. C = -1 => no
// control. Constant matrix entries fold after inlining (CNOT -> selects,
// RZ -> phase mults, RY -> real math).
template<int C, int T>
__device__ __forceinline__ void applyGate(
    float pr[8], float pi[8],
    float u00r, float u00i, float u01r, float u01i,
    float u10r, float u10i, float u11r, float u11i, int lane)
{
  if constexpr (T <= 4) {
    // target on a lane wire: exchange partner amplitudes across lanes
    constexpr int lb = 4 - T;
    constexpr int mask = 1 << lb;
    const bool hi = (lane >> lb) & 1;   // my side of the target bit
    bool lctrl = true;
    if constexpr (C >= 0 && C <= 4) lctrl = (lane >> (4 - C)) & 1;
#pragma unroll
    for (int i = 0; i < 8; ++i) {
      if constexpr (C >= 5) { if (((i >> (7 - C)) & 1) == 0) continue; }
      float qr = swzx<mask>(pr[i]);
      float qi = swzx<mask>(pi[i]);
      // alpha = amplitude with target bit 0, beta = target bit 1
      float ar = hi ? u10r : u00r, ai = hi ? u10i : u00i;  // coeff * alpha
      float br = hi ? u11r : u01r, bi = hi ? u11i : u01i;  // coeff * beta
      float a0r = hi ? qr : pr[i], a0i = hi ? qi : pi[i];
      float a1r = hi ? pr[i] : qr, a1i = hi ? pi[i] : qi;
      float nr = ar*a0r - ai*a0i + br*a1r - bi*a1i;
      float ni = ar*a0i + ai*a0r + br*a1i + bi*a1r;
      if constexpr (C >= 0 && C <= 4) {
        pr[i] = lctrl ? nr : pr[i];
        pi[i] = lctrl ? ni : pi[i];
      } else { pr[i] = nr; pi[i] = ni; }
    }
  } else {
    // target on a local wire: pure in-register pair updates
    constexpr int tb = 7 - T;
    bool lctrl = true;
    if constexpr (C >= 0 && C <= 4) lctrl = (lane >> (4 - C)) & 1;
#pragma unroll
    for (int i = 0; i < 8; ++i) {
      if ((i >> tb) & 1) continue;  // compile-time after unroll
      if constexpr (C >= 5) { if (((i >> (7 - C)) & 1) == 0) continue; }
      const int j = i | (1 << tb);
      float a0r = pr[i], a0i = pi[i], a1r = pr[j], a1i = pi[j];
      float nr0 = u00r*a0r - u00i*a0i + u01r*a1r - u01i*a1i;
      float ni0 = u00r*a0i + u00i*a0r + u01r*a1i + u01i*a1r;
      float nr1 = u10r*a0r - u10i*a0i + u11r*a1r - u11i*a1i;
      float ni1 = u10r*a0i + u10i*a0r + u11r*a1i + u11i*a1r;
      if constexpr (C >= 0 && C <= 4) {
        pr[i] = lctrl ? nr0 : pr[i]; pi[i] = lctrl ? ni0 : pi[i];
        pr[j] = lctrl ? nr1 : pr[j]; pi[j] = lctrl ? ni1 : pi[j];
      } else { pr[i] = nr0; pi[i] = ni0; pr[j] = nr1; pi[j] = ni1; }
    }
  }
}

// full-wave butterfly sum (compile-time swizzle patterns)
__device__ __forceinline__ float waveSum(float v) {
  v += swzx<16>(v); v += swzx<8>(v); v += swzx<4>(v);
  v += swzx<2>(v);  v += swzx<1>(v);
  return v;
}

__global__ __launch_bounds__(256) void qsim_kernel(
    const float* __restrict__ theta, const float* __restrict__ phi,
    const float* __restrict__ weights, const float* __restrict__ u3p,
    float* __restrict__ feats, int B)
{
  const int lane = threadIdx.x & 31;
  const int elem = blockIdx.x * (blockDim.x >> 5) + (threadIdx.x >> 5);
  if (elem >= B) return;

  float pr[8], pi[8];
#pragma unroll
  for (int i = 0; i < 8; ++i) { pr[i] = 0.f; pi[i] = 0.f; }
  pr[0] = (lane == 0) ? 1.f : 0.f;   // |0...0>

  const float th = theta[elem], ph = phi[elem];
  const float ct = cosf(0.5f * th), st = sinf(0.5f * th);
  const float cp = cosf(0.5f * ph), sp = sinf(0.5f * ph);

  // ---- prep_state: 4 cycles of per-wire RY(theta),RZ(phi) then CNOT ring ----
  for (int cyc = 0; cyc < 4; ++cyc) {
#define PREPW(W) \
    applyGate<-1,W>(pr,pi, ct,0.f, -st,0.f, st,0.f, ct,0.f, lane); \
    applyGate<-1,W>(pr,pi, cp,-sp, 0.f,0.f, 0.f,0.f, cp,sp, lane);
    PREPW(0) PREPW(1) PREPW(2) PREPW(3) PREPW(4) PREPW(5) PREPW(6) PREPW(7)
#undef PREPW
#define CNOT(Cc,Tt) applyGate<Cc,Tt>(pr,pi, 0.f,0.f, 1.f,0.f, 1.f,0.f, 0.f,0.f, lane);
    CNOT(0,1) CNOT(1,2) CNOT(2,3) CNOT(3,4) CNOT(4,5) CNOT(5,6) CNOT(6,7) CNOT(7,0)
#undef CNOT
  }

  // save prepared state (re-used by all three branches)
  float sr[8], si[8];
#pragma unroll
  for (int i = 0; i < 8; ++i) { sr[i] = pr[i]; si[i] = pi[i]; }

  // cos/sin of the 10 shared controlled-rotation weights
  float cw[10], swn[10];
#pragma unroll
  for (int g = 0; g < 10; ++g) {
    float w = 0.5f * weights[g];
    cw[g] = cosf(w); swn[g] = sinf(w);
  }

  // ---- three branches: b=0 CRX, b=1 CRY, b=2 CRZ ----
  for (int b = 0; b < 3; ++b) {
#pragma unroll
    for (int i = 0; i < 8; ++i) { pr[i] = sr[i]; pi[i] = si[i]; }
    const bool isX = (b == 0), isY = (b == 1), isZ = (b == 2);

    // controlled rotation, matrix selected per branch:
    // RX: [[c,-is],[-is,c]]  RY: [[c,-s],[s,c]]  RZ: diag(c-is, c+is)
#define CR(G,Cc,Tt) { float cg = cw[G], sg = swn[G];                 \
      float z00i = isZ ? -sg : 0.f;                                  \
      float z01r = isY ? -sg : 0.f; float z01i = isX ? -sg : 0.f;    \
      float z10r = isY ?  sg : 0.f; float z10i = isX ? -sg : 0.f;    \
      float z11i = isZ ?  sg : 0.f;                                  \
      applyGate<Cc,Tt>(pr,pi, cg,z00i, z01r,z01i, z10r,z10i, cg,z11i, lane); }

    // U3(t,p,l) = [[cos(t/2), -e^{il} sin(t/2)], [e^{ip} sin(t/2), e^{i(p+l)} cos(t/2)]]
#define U3G(J,W) { const float* q = u3p + (6*b + (J))*3;             \
      float tt = q[0], pp = q[1], ll = q[2];                         \
      float c2 = cosf(0.5f*tt), s2 = sinf(0.5f*tt);                  \
      float clr = cosf(ll),  sli = sinf(ll);                         \
      float cpr = cosf(pp),  spi = sinf(pp);                         \
      float cplr = cosf(pp+ll), spli = sinf(pp+ll);                  \
      applyGate<-1,W>(pr,pi, c2,0.f, -clr*s2,-sli*s2,                \
                      cpr*s2,spi*s2, cplr*c2,spli*c2, lane); }

    // CR_PAIRS_1 with weights[0..6]
    CR(0,0,1) CR(1,2,3) CR(2,4,5) CR(3,6,7) CR(4,1,2) CR(5,3,4) CR(6,5,6)
    // U3 on wires 1,3,5,7 (rows 0..3 of this branch's u3 block)
    U3G(0,1) U3G(1,3) U3G(2,5) U3G(3,7)
    // CR_PAIRS_2 with weights[7..9]
    CR(7,1,3) CR(8,5,7) CR(9,3,5)
    // U3 on wires 3,7 (rows 4,5)
    U3G(4,3) U3G(5,7)
#undef CR
#undef U3G

    // expval_z wire 3 (lane bit 1) and wire 7 (local bit 0)
    float e3 = 0.f, e7 = 0.f;
#pragma unroll
    for (int i = 0; i < 8; ++i) {
      float m = pr[i]*pr[i] + pi[i]*pi[i];
      e3 += m;
      e7 += (i & 1) ? -m : m;
    }
    e3 = ((lane >> 1) & 1) ? -e3 : e3;
    e3 = waveSum(e3);
    e7 = waveSum(e7);
    if (lane == 0) {
      feats[(size_t)elem * 6 + 2*b + 0] = e3;
      feats[(size_t)elem * 6 + 2*b + 1] = e7;
    }
  }
}

// -----------------------------------------------------------------------------
// MLP head via V_WMMA_F32_16X16X4_F32: hidden(16x15) = feats(16x6) @ W1(6x15)
// K=6 split into two K=4 WMMAs (second chunk zero-padded).
// A layout (16x4 f32): lanes 0-15 hold M=lane, K=0..1; lanes 16-31 K=2..3.
// B layout (4x16 f32): lanes 0-15 hold N=lane, K=0..1; lanes 16-31 K=2..3.
// D layout (16x16 f32): vgpr r = row (r, or 8+r in high half), col = lane&15.
// All edge masking is branchless (clamped index + multiply) so no exec-mask
// churn / serialized divergent loads.
// -----------------------------------------------------------------------------
__global__ __launch_bounds__(256) void mlp_kernel(
    const float* __restrict__ feats,
    const float* __restrict__ W1, const float* __restrict__ b1,
    const float* __restrict__ W2, const float* __restrict__ b2,
    float* __restrict__ out, int B)
{
  const int lane = threadIdx.x & 31;
  const int wave = blockIdx.x * (blockDim.x >> 5) + (threadIdx.x >> 5);
  const int r0 = wave * 16;
  if (r0 >= B) return;          // wave-uniform; EXEC stays all-ones for WMMA
  const int n = lane & 15;
  const bool hiHalf = lane >= 16;
  const int nc = (n < 15) ? n : 14;          // clamped column index
  const float nm = (n < 15) ? 1.f : 0.f;     // column-15 zero mask

  const float* fr = feats + (size_t)(r0 + n) * 6;
  v2f a0, a1;
  if (!hiHalf) { a0[0] = fr[0]; a0[1] = fr[1]; a1[0] = fr[4]; a1[1] = fr[5]; }
  else         { a0[0] = fr[2]; a0[1] = fr[3]; a1[0] = 0.f;   a1[1] = 0.f;   }

  const int kb = hiHalf ? 2 : 0;
  v2f bv0, bv1;
  bv0[0] = W1[(kb + 0) * 15 + nc] * nm;
  bv0[1] = W1[(kb + 1) * 15 + nc] * nm;
  bv1[0] = hiHalf ? 0.f : W1[4 * 15 + nc] * nm;
  bv1[1] = hiHalf ? 0.f : W1[5 * 15 + nc] * nm;

  v8f acc = {};
  acc = __builtin_amdgcn_wmma_f32_16x16x4_f32(false, a0, false, bv0,
                                              (short)0, acc, false, false);
  acc = __builtin_amdgcn_wmma_f32_16x16x4_f32(false, a1, false, bv1,
                                              (short)0, acc, false, false);

  const float bias = b1[nc];                 // col 15 masked via w2n
  const float w2n  = W2[nc] * nm;
  float rsum[8];
#pragma unroll
  for (int i = 0; i < 8; ++i) {
    float h = tanhf(acc[i] + bias);
    float v = h * w2n;
    v += swzx<1>(v); v += swzx<2>(v); v += swzx<4>(v); v += swzx<8>(v);
    rsum[i] = v;                 // lanes with n==0 now hold the row dot product
  }
  if (n == 0) {
    const int rb = r0 + (hiHalf ? 8 : 0);
    const float bb = b2[0];
#pragma unroll
    for (int i = 0; i < 8; ++i) {
      float z = rsum[i] + bb;
      out[rb + i] = 1.f / (1.f + expf(-z));
    }
  }
}

extern "C" void kernel_launch(void* const* d_in, const int* in_sizes, int n_in,
                              void* d_out, int out_size, void* d_ws, size_t ws_size,
                              hipStream_t stream) {
  const float* theta = (const float*)d_in[0];
  const float* phi   = (const float*)d_in[1];
  const float* w     = (const float*)d_in[2];
  const float* u3p   = (const float*)d_in[3];
  const float* W1    = (const float*)d_in[4];
  const float* b1    = (const float*)d_in[5];
  const float* W2    = (const float*)d_in[6];
  const float* b2    = (const float*)d_in[7];
  float* out   = (float*)d_out;
  float* feats = (float*)d_ws;            // B*6 floats = 1.5 MB scratch
  const int B = in_sizes[0];

  const int wavesPerBlock = 256 / 32;
  dim3 blk(256);
  dim3 g1((B + wavesPerBlock - 1) / wavesPerBlock);   // one wave per element
  hipLaunchKernelGGL(qsim_kernel, g1, blk, 0, stream, theta, phi, w, u3p, feats, B);

  const int tiles = (B + 15) / 16;                    // one wave per 16 rows
  dim3 g2((tiles + wavesPerBlock - 1) / wavesPerBlock);
  hipLaunchKernelGGL(mlp_kernel, g2, blk, 0, stream, feats, W1, b1, W2, b2, out, B);
}